// GCN_60790967107992
// MI455X (gfx1250) — compile-verified
//
#include <hip/hip_runtime.h>
#include <math.h>

// ---------------------------------------------------------------------------
// 2-layer GCN for MI455X (gfx1250, wave32).
//  - GEMMs via V_WMMA_F32_16X16X4_F32, fully-unrolled K loop, contiguous b64
//    fragment loads (A row-major, B pre-transposed + zero-padded).
//  - Edge gather/scatter is the hot loop; per-layer working set (~51 MB)
//    fits in the 192 MB L2, so f32 atomics land in L2, not HBM.
// ---------------------------------------------------------------------------

typedef float v2f __attribute__((ext_vector_type(2)));
typedef float v8f __attribute__((ext_vector_type(8)));

#define WAVE 32

// ---------------- degree / normalization ----------------
__global__ void k_init_deg(float* __restrict__ deg, int n) {
  int i = blockIdx.x * blockDim.x + threadIdx.x;
  if (i < n) deg[i] = 1.0f;                       // self-loop contributes 1
}

__global__ void k_count_deg(const int* __restrict__ dst, float* __restrict__ deg, int E) {
  int e = blockIdx.x * blockDim.x + threadIdx.x;
  if (e < E) atomicAdd(&deg[dst[e]], 1.0f);       // global_atomic_add_f32 (no-return)
}

__global__ void k_deg_to_dinv(float* __restrict__ deg, int n) {
  int i = blockIdx.x * blockDim.x + threadIdx.x;
  if (i < n) {
    float d = deg[i];
    deg[i] = (d > 0.0f) ? rsqrtf(d) : 0.0f;
  }
}

// ---------------- weight transpose + zero-pad: Wt[n*K + k] = W[k*NB + n] ----
// Wt is [NBPAD x K] so each output column's K-vector is contiguous (b64 loads
// in the GEMM) and columns NB..NBPAD-1 are zero (no guards in the hot loop).
__global__ void k_transpose_pad(const float* __restrict__ W, float* __restrict__ Wt,
                                int K, int NB, int NBPAD) {
  int idx = blockIdx.x * blockDim.x + threadIdx.x;
  if (idx >= NBPAD * K) return;
  int n = idx / K;
  int k = idx % K;
  Wt[idx] = (n < NB) ? W[k * NB + n] : 0.0f;
}

// ---------------- fp32 WMMA GEMM: C[M x NB] = A[M x K] @ B[K x NB] ----------
// One wave per 16x16 output tile; K stepped by 4 (V_WMMA_F32_16X16X4_F32).
// ISA layout (7.12.2):
//   A 16x4 : lane&15 = M, VGPR j holds K = 2*(lane>>4) + j
//   B 4x16 : lane&15 = N, VGPR j holds K = 2*(lane>>4) + j
//   C 16x16: VGPR r holds row M = r + 8*(lane>>4), col N = lane&15
// Bt is the transposed+padded weight buffer [NBPAD x K].
template <int K, int NB, int NBPAD>
__global__ void k_gemm_wmma_f32(const float* __restrict__ A,
                                const float* __restrict__ Bt,
                                float* __restrict__ C, int M) {
  constexpr int NTILES = NBPAD / 16;
  const int w = blockIdx.x * (blockDim.x / WAVE) + (threadIdx.x / WAVE);
  const int Mtiles = (M + 15) >> 4;
  if (w >= Mtiles * NTILES) return;               // wave-uniform: EXEC stays all-1s
  const int mt = w / NTILES;
  const int nt = w % NTILES;
  const int lane = threadIdx.x & (WAVE - 1);
  const int half = lane >> 4;                     // 0: lanes 0-15, 1: lanes 16-31
  const int mr = lane & 15;
  // Clamped A row: a duplicated row only feeds output rows >= M, never stored.
  const int m = min(mt * 16 + mr, M - 1);
  const int n = nt * 16 + mr;                     // B/C column (zeros if n >= NB)

  const float* Ap = A + (size_t)m * K + 2 * half;   // even offset -> 8B aligned
  const float* Bp = Bt + (size_t)n * K + 2 * half;  // contiguous column of B

  v8f acc = {0.f, 0.f, 0.f, 0.f, 0.f, 0.f, 0.f, 0.f};

#pragma unroll
  for (int kk = 0; kk < K; kk += 4) {
    const v2f a = *reinterpret_cast<const v2f*>(Ap + kk);
    const v2f b = *reinterpret_cast<const v2f*>(Bp + kk);
    acc = __builtin_amdgcn_wmma_f32_16x16x4_f32(
        /*neg_a=*/false, a, /*neg_b=*/false, b,
        /*c_mod=*/(short)0, acc, /*reuse_a=*/false, /*reuse_b=*/false);
  }

  if (n < NB) {
#pragma unroll
    for (int r = 0; r < 8; ++r) {
      const int mo = mt * 16 + r + 8 * half;
      if (mo < M) C[(size_t)mo * NB + n] = acc[r];
    }
  }
}

// ---------------- edge scatter: agg[dst] += norm * h[src] -------------------
// One thread per (edge, 4-feature chunk); float4 gather + 4 f32 atomics.
template <int F>
__global__ void k_scatter(const float* __restrict__ h,
                          const float* __restrict__ dinv,
                          const int* __restrict__ src,
                          const int* __restrict__ dst,
                          float* __restrict__ agg, int E) {
  constexpr int CH = F / 4;
  const long long t = (long long)blockIdx.x * blockDim.x + threadIdx.x;
  if (t >= (long long)E * CH) return;
  const int e = (int)(t / CH);                    // constant divisor
  const int c = (int)(t % CH);
  const int s = src[e];
  const int d = dst[e];
  const float wgt = dinv[s] * dinv[d];
  const float4 hv = *reinterpret_cast<const float4*>(h + (size_t)s * F + 4 * c);
  float* o = agg + (size_t)d * F + 4 * c;
  atomicAdd(o + 0, hv.x * wgt);
  atomicAdd(o + 1, hv.y * wgt);
  atomicAdd(o + 2, hv.z * wgt);
  atomicAdd(o + 3, hv.w * wgt);
}

// ---------------- layer-1 epilogue: + self-loop + bias, ReLU (in place) -----
__global__ void k_post_relu(float* __restrict__ agg,
                            const float* __restrict__ h,
                            const float* __restrict__ dinv,
                            const float* __restrict__ bias,
                            int N, int F) {
  const int idx = blockIdx.x * blockDim.x + threadIdx.x;
  if (idx >= N * F) return;
  const int i = idx / F;
  const int f = idx % F;
  const float di = dinv[i];
  const float v = agg[idx] + di * di * h[idx] + bias[f];
  agg[idx] = fmaxf(v, 0.0f);
}

// ---------------- layer-2 epilogue + log_softmax (one wave per node) --------
__global__ void k_post_logsoftmax(const float* __restrict__ agg,
                                  const float* __restrict__ h,
                                  const float* __restrict__ dinv,
                                  const float* __restrict__ bias,
                                  float* __restrict__ out,
                                  int N, int C) {
  const int wavesPerBlock = blockDim.x / WAVE;
  const int node = blockIdx.x * wavesPerBlock + (threadIdx.x / WAVE);
  if (node >= N) return;                          // wave-uniform exit
  const int lane = threadIdx.x & (WAVE - 1);
  const float di = dinv[node];
  const float w = di * di;
  const float* ar = agg + (size_t)node * C;
  const float* hr = h + (size_t)node * C;

  const int f0 = lane;
  const int f1 = lane + WAVE;
  float v0 = -INFINITY, v1 = -INFINITY;
  if (f0 < C) v0 = ar[f0] + w * hr[f0] + bias[f0];
  if (f1 < C) v1 = ar[f1] + w * hr[f1] + bias[f1];

  // wave32 max reduction
  float mx = fmaxf(v0, v1);
#pragma unroll
  for (int off = 16; off > 0; off >>= 1)
    mx = fmaxf(mx, __shfl_xor(mx, off, WAVE));

  // wave32 sum of exp
  float s = 0.0f;
  if (f0 < C) s += expf(v0 - mx);
  if (f1 < C) s += expf(v1 - mx);
#pragma unroll
  for (int off = 16; off > 0; off >>= 1)
    s += __shfl_xor(s, off, WAVE);

  const float ls = mx + logf(s);
  if (f0 < C) out[(size_t)node * C + f0] = v0 - ls;
  if (f1 < C) out[(size_t)node * C + f1] = v1 - ls;
}

// ---------------------------------------------------------------------------
extern "C" void kernel_launch(void* const* d_in, const int* in_sizes, int n_in,
                              void* d_out, int out_size, void* d_ws, size_t ws_size,
                              hipStream_t stream) {
  const float* x  = (const float*)d_in[0];   // [N,128]
  const int*   ei = (const int*)d_in[1];     // [2,E] (JAX default int32)
  const float* W1 = (const float*)d_in[2];   // [128,64]
  const float* b1 = (const float*)d_in[3];   // [64]
  const float* W2 = (const float*)d_in[4];   // [64,40]
  const float* b2 = (const float*)d_in[5];   // [40]

  constexpr int F  = 128;                    // NUM_FEATURES
  constexpr int H  = 64;                     // HIDDEN
  constexpr int C  = 40;                     // NUM_CLASSES
  constexpr int CP = 48;                     // C padded to 16-multiple
  const int N = in_sizes[0] / F;             // 100000
  const int E = in_sizes[1] / 2;             // 1600000
  const int* src = ei;
  const int* dst = ei + E;

  // workspace layout (floats); every region starts 16B-aligned (N*4, N*H*4,
  // N*C*4 are all multiples of 16 bytes for N=100000).
  float* ws   = (float*)d_ws;
  float* dinv = ws;                               // [N]   (deg, then rsqrt in place)
  float* h1   = dinv + N;                         // [N*H]
  float* agg1 = h1 + (size_t)N * H;               // [N*H]
  float* h2   = agg1 + (size_t)N * H;             // [N*C]
  float* agg2 = h2 + (size_t)N * C;               // [N*C]
  float* W1t  = agg2 + (size_t)N * C;             // [H  * F]  = 8192
  float* W2t  = W1t + (size_t)H * F;              // [CP * H]  = 3072

  hipMemsetAsync(agg1, 0, (size_t)N * H * sizeof(float), stream);
  hipMemsetAsync(agg2, 0, (size_t)N * C * sizeof(float), stream);

  // normalization: deg (with self-loop) -> dinv = deg^{-1/2}
  k_init_deg<<<(N + 255) / 256, 256, 0, stream>>>(dinv, N);
  k_count_deg<<<(E + 255) / 256, 256, 0, stream>>>(dst, dinv, E);
  k_deg_to_dinv<<<(N + 255) / 256, 256, 0, stream>>>(dinv, N);

  // weight transposes (tiny)
  k_transpose_pad<<<(H * F + 255) / 256, 256, 0, stream>>>(W1, W1t, F, H, H);
  k_transpose_pad<<<(CP * H + 255) / 256, 256, 0, stream>>>(W2, W2t, H, C, CP);

  // layer 1: h1 = x @ W1  (WMMA fp32, 16x16 tiles, K=128 fully unrolled)
  {
    const int tiles = ((N + 15) / 16) * (H / 16);  // 25000 waves
    k_gemm_wmma_f32<F, H, H><<<(tiles + 7) / 8, 256, 0, stream>>>(x, W1t, h1, N);
  }
  // scatter: agg1[dst] += norm * h1[src]
  {
    const long long total = (long long)E * (H / 4);
    k_scatter<H><<<(unsigned)((total + 255) / 256), 256, 0, stream>>>(
        h1, dinv, src, dst, agg1, E);
  }
  // + self-loop + b1, ReLU (in place in agg1)
  k_post_relu<<<((size_t)N * H + 255) / 256, 256, 0, stream>>>(agg1, h1, dinv, b1, N, H);

  // layer 2: h2 = relu(agg1) @ W2 (K=64 fully unrolled, N padded 40->48)
  {
    const int tiles = ((N + 15) / 16) * (CP / 16); // 18750 waves
    k_gemm_wmma_f32<H, C, CP><<<(tiles + 7) / 8, 256, 0, stream>>>(agg1, W2t, h2, N);
  }
  // scatter: agg2[dst] += norm * h2[src]
  {
    const long long total = (long long)E * (C / 4);
    k_scatter<C><<<(unsigned)((total + 255) / 256), 256, 0, stream>>>(
        h2, dinv, src, dst, agg2, E);
  }
  // + self-loop + b2, log_softmax -> d_out
  k_post_logsoftmax<<<(N + 7) / 8, 256, 0, stream>>>(agg2, h2, dinv, b2,
                                                     (float*)d_out, N, C);
}